// MultiQueryAttention_16690242912480
// MI455X (gfx1250) — compile-verified
//
#include <hip/hip_runtime.h>
#include <hip/hip_bf16.h>

// ---------------------------------------------------------------------------
// MQA forward for MI455X (gfx1250), wave32, f32 WMMA 16x16x4 + TDM async tiles.
//  B=2, S=2048, D_MODEL=1024, H=16, DK=64.
// Roofline: ~70 GFLOP f32 (with flash recompute) vs mandatory 537 MB softmax
// weights store -> memory bound (~23 us floor at 23.3 TB/s). Design goal:
// touch the big tensor exactly once, keep everything else L2-resident,
// and feed WMMA from LDS via the Tensor Data Mover.
// ---------------------------------------------------------------------------

typedef __attribute__((ext_vector_type(2))) float    v2f;
typedef __attribute__((ext_vector_type(8))) float    v8f;
typedef __attribute__((ext_vector_type(4))) unsigned v4u;
typedef __attribute__((ext_vector_type(4))) int      v4i;
typedef __attribute__((ext_vector_type(8))) int      v8i;

#define S_LEN   2048
#define DMODEL  1024
#define NHEAD   16
#define DK      64
#define NB      2
#define NROWS   (NB * S_LEN)

__device__ __forceinline__ v8f wmma4(v2f a, v2f b, v8f c) {
  // V_WMMA_F32_16X16X4_F32 : D = A(16x4) * B(4x16) + C(16x16)
  return __builtin_amdgcn_wmma_f32_16x16x4_f32(
      false, a, false, b, (short)0, c, false, false);
}

// ---------------------------------------------------------------------------
// TDM: build a 2-D tile descriptor (D#, ISA 8.3/8.4) and issue
// TENSOR_LOAD_TO_LDS. data_size=4B. Optional LDS padding: 2 DWORDs of pad
// every 32 DWORDs (-> LDS row stride 34 floats for our 32-float tile rows).
// Must be called wave-uniformly (EXEC ignored by TDM; tracked by TENSORcnt).
// ---------------------------------------------------------------------------
__device__ __forceinline__ unsigned lds_off(const void* p) {
  // addrspace(3) objects: low 32 bits of the flat address are the LDS offset
  return (unsigned)(size_t)p;
}

__device__ __forceinline__ void tdm_load_2d(unsigned ldsaddr, const float* gptr,
                                            unsigned tile_cols, unsigned tile_rows,
                                            unsigned row_stride_elems, int pad_en) {
  const unsigned long long ga = (unsigned long long)(size_t)gptr;
  v4u g0;
  g0[0] = 1u;                                            // count=1, user mode
  g0[1] = ldsaddr;                                       // lds_addr (bytes)
  g0[2] = (unsigned)(ga & 0xFFFFFFFFu);                  // global_addr[31:0]
  g0[3] = (unsigned)((ga >> 32) & 0x01FFFFFFu) | (2u << 30);  // [56:32] | type=2

  const unsigned td0 = row_stride_elems;                 // tensor_dim0 (elems)
  const unsigned td1 = 0x00100000u;                      // tensor_dim1: generous
  const unsigned long long st0 = (unsigned long long)row_stride_elems;
  v8i g1;
  g1[0] = (int)((2u << 16) |                             // data_size = 4B
                ((unsigned)(pad_en ? 1u : 0u) << 20) |   // pad_enable
                (4u << 22) |                             // pad_interval: 32 DW
                (1u << 25));                             // pad_amount: 2 DW
  g1[1] = (int)((td0 & 0xFFFFu) << 16);                  // [15:0] atomic_bar=0
  g1[2] = (int)(((td0 >> 16) & 0xFFFFu) | ((td1 & 0xFFFFu) << 16));
  g1[3] = (int)(((td1 >> 16) & 0xFFFFu) | (tile_cols << 16));   // tile_dim0
  g1[4] = (int)(tile_rows & 0xFFFFu);                    // tile_dim1; dim2=0
  g1[5] = (int)(st0 & 0xFFFFFFFFu);                      // dim0_stride lo
  g1[6] = (int)((st0 >> 32) & 0xFFFFu);                  // dim0_stride hi
  g1[7] = 0;
  const v4i gz4 = {0, 0, 0, 0};                          // 2-D: groups 2/3 off
  const v8i gz8 = {0, 0, 0, 0, 0, 0, 0, 0};
  __builtin_amdgcn_tensor_load_to_lds(g0, g1, gz4, gz4, gz8, 0);
}

// ---------------------------------------------------------------------------
// C[M,N] = A[M,K] * W[N,K]^T + bias[N]
// 256 thr = 8 waves; block tile 128(M) x 64(N), K-step 32, TDM double buffer.
// Wave w owns rows [w*16, w*16+16); 4 N-tiles; fragments read from padded LDS
// (stride 34 -> 16 distinct banks across fragment rows).
// ---------------------------------------------------------------------------
#define GK    32
#define T_LD  34

__global__ __launch_bounds__(256) void gemm_bias_kernel(
    const float* __restrict__ A, const float* __restrict__ W,
    const float* __restrict__ bias, float* __restrict__ C,
    int M, int N, int K) {
  __shared__ float At[2][128 * T_LD];   // 34.8 KB
  __shared__ float Wt[2][64 * T_LD];    // 17.4 KB

  const int wid  = threadIdx.x >> 5;
  const int lane = threadIdx.x & 31;
  const int row  = lane & 15;
  const int half = lane >> 4;
  const int m0   = blockIdx.x * 128;
  const int n0   = blockIdx.y * 64;

  if (wid == 0) {
    tdm_load_2d(lds_off(&At[0][0]), A + (size_t)m0 * K, GK, 128, (unsigned)K, 1);
    tdm_load_2d(lds_off(&Wt[0][0]), W + (size_t)n0 * K, GK, 64,  (unsigned)K, 1);
    __builtin_amdgcn_s_wait_tensorcnt(0);
  }
  __syncthreads();

  v8f acc[4] = {};
  const int nk = K / GK;
  for (int t = 0; t < nk; ++t) {
    const int cur = t & 1;
    const int nxt = cur ^ 1;
    if (wid == 0 && t + 1 < nk) {       // kick next tile while computing
      tdm_load_2d(lds_off(&At[nxt][0]), A + (size_t)m0 * K + (t + 1) * GK,
                  GK, 128, (unsigned)K, 1);
      tdm_load_2d(lds_off(&Wt[nxt][0]), W + (size_t)n0 * K + (t + 1) * GK,
                  GK, 64, (unsigned)K, 1);
    }
    const float* at = &At[cur][(wid * 16 + row) * T_LD];
    const float* wt = &Wt[cur][row * T_LD];
#pragma unroll
    for (int ks = 0; ks < GK / 4; ++ks) {
      const int kk = ks * 4 + 2 * half;
      v2f a; a.x = at[kk]; a.y = at[kk + 1];
#pragma unroll
      for (int j = 0; j < 4; ++j) {
        v2f bf; bf.x = wt[j * 16 * T_LD + kk]; bf.y = wt[j * 16 * T_LD + kk + 1];
        acc[j] = wmma4(a, bf, acc[j]);
      }
    }
    if (wid == 0 && t + 1 < nk) __builtin_amdgcn_s_wait_tensorcnt(0);
    __syncthreads();
  }

#pragma unroll
  for (int j = 0; j < 4; ++j) {
    const int col = n0 + j * 16 + row;
    const float bv = bias[col];
#pragma unroll
    for (int i = 0; i < 8; ++i) {
      const int m = m0 + wid * 16 + i + 8 * half;
      C[(size_t)m * N + col] = acc[j][i] + bv;
    }
  }
}

// ---------------------------------------------------------------------------
// Attention: one block = 16 query rows of one (b,h). Flash-style two passes
// with recompute (logits GEMM is ~free vs the 537 MB weights stream).
// ---------------------------------------------------------------------------
#define CHUNK  512
#define CH_LD  516   // (4r + c) % 64 distinct across rows; rows 16B-aligned
#define QS_LD  66

__device__ __forceinline__ void compute_logits_chunk(
    const float* __restrict__ qs, const float* __restrict__ Kb,
    float* __restrict__ ch, int c0, int wid, int row, int half) {
  const float* krow[4];
#pragma unroll
  for (int j = 0; j < 4; ++j)
    krow[j] = Kb + (size_t)(c0 + (wid * 4 + j) * 16 + row) * (2 * DK);

  v8f acc[4] = {};
  v2f b[4], bn[4];
#pragma unroll
  for (int j = 0; j < 4; ++j) {        // prefetch k-step 0
    b[j].x = krow[j][2 * half]; b[j].y = krow[j][2 * half + 1];
  }
#pragma unroll 4
  for (int ks = 0; ks < 16; ++ks) {    // K = 64, step 4, pipelined depth 1
    const int kk = ks * 4 + 2 * half;
    if (ks < 15) {
#pragma unroll
      for (int j = 0; j < 4; ++j) {
        bn[j].x = krow[j][kk + 4]; bn[j].y = krow[j][kk + 5];
      }
    }
    v2f a; a.x = qs[row * QS_LD + kk]; a.y = qs[row * QS_LD + kk + 1];
#pragma unroll
    for (int j = 0; j < 4; ++j) acc[j] = wmma4(a, b[j], acc[j]);
#pragma unroll
    for (int j = 0; j < 4; ++j) b[j] = bn[j];
  }
#pragma unroll
  for (int j = 0; j < 4; ++j) {
    const int cl = (wid * 4 + j) * 16 + row;
#pragma unroll
    for (int i = 0; i < 8; ++i) ch[(i + 8 * half) * CH_LD + cl] = acc[j][i];
  }
}

__global__ __launch_bounds__(256) void mqa_attn_kernel(
    const float* __restrict__ Qp, const float* __restrict__ KV,
    float* __restrict__ weights_out, float* __restrict__ attn_out) {
  __shared__ __attribute__((aligned(16))) float ch[16 * CH_LD];  // 33 KB
  __shared__ float qs[16 * QS_LD];
  __shared__ float red[16 * 16];
  __shared__ float rowmax[16];
  __shared__ float rowsum[16];
  __shared__ float part[2 * 16 * DK];

  const int tid  = threadIdx.x;
  const int wid  = tid >> 5;
  const int lane = tid & 31;
  const int row  = lane & 15;
  const int half = lane >> 4;

  const int qblk = blockIdx.x & 127;
  const int bh   = blockIdx.x >> 7;
  const int h    = bh & 15;
  const int b    = bh >> 4;
  const int q0   = qblk * 16;

  for (int e = tid; e < 16 * DK; e += 256) {
    const int r = e >> 6, c = e & 63;
    qs[r * QS_LD + c] =
        Qp[(size_t)(b * S_LEN + q0 + r) * DMODEL + h * DK + c] * 0.125f;
  }
  if (tid < 16) { rowmax[tid] = -3.0e38f; rowsum[tid] = 0.0f; }
  __syncthreads();

  const float* Kb = KV + (size_t)b * S_LEN * (2 * DK);
  const float* Vb = Kb + DK;

  // ------------------ pass 1: online row max + exp-sum ------------------
  for (int c0 = 0; c0 < S_LEN; c0 += CHUNK) {
    compute_logits_chunk(qs, Kb, ch, c0, wid, row, half);
    __syncthreads();
    {
      const int r = tid >> 4, j = tid & 15;
      float m = -3.0e38f;
      for (int c = j; c < CHUNK; c += 16) m = fmaxf(m, ch[r * CH_LD + c]);
      red[r * 16 + j] = m;
    }
    __syncthreads();
    if (tid < 16) {
      float cm = red[tid * 16];
      for (int j = 1; j < 16; ++j) cm = fmaxf(cm, red[tid * 16 + j]);
      const float om = rowmax[tid];
      const float nm = fmaxf(om, cm);
      rowsum[tid] *= __expf(om - nm);
      rowmax[tid] = nm;
    }
    __syncthreads();
    {
      const int r = tid >> 4, j = tid & 15;
      const float mx = rowmax[r];
      float s = 0.0f;
      for (int c = j; c < CHUNK; c += 16) s += __expf(ch[r * CH_LD + c] - mx);
      red[r * 16 + j] = s;
    }
    __syncthreads();
    if (tid < 16) {
      float s = 0.0f;
      for (int j = 0; j < 16; ++j) s += red[tid * 16 + j];
      rowsum[tid] += s;
    }
    __syncthreads();
  }

  // -------- pass 2: recompute, normalize, emit weights, attn GEMM --------
  v8f aacc = {};
  const int dtile  = wid & 3;
  const int khalf2 = wid >> 2;
  const int dcol   = dtile * 16 + row;
  float* Wout = weights_out + (((size_t)(b * NHEAD + h)) * S_LEN + q0) * S_LEN;

  for (int c0 = 0; c0 < S_LEN; c0 += CHUNK) {
    compute_logits_chunk(qs, Kb, ch, c0, wid, row, half);
    __syncthreads();
    for (int e = tid; e < 16 * CHUNK; e += 256) {     // normalize in LDS
      const int r = e >> 9, c = e & (CHUNK - 1);
      ch[r * CH_LD + c] = __expf(ch[r * CH_LD + c] - rowmax[r]) / rowsum[r];
    }
    __syncthreads();
    // stream final weights to HBM as b128 stores (dominant traffic, once)
    for (int e = tid; e < 16 * (CHUNK / 4); e += 256) {
      const int r = e >> 7, c4 = (e & 127) * 4;
      const float4 v = *(const float4*)&ch[r * CH_LD + c4];
      *(float4*)&Wout[(size_t)r * S_LEN + c0 + c4] = v;
    }
    // attn partial: wave (dtile, khalf2), 64 pipelined WMMA k-steps
    v2f bf, bfn;
    {
      const float* vr = Vb + (size_t)(c0 + khalf2 * 256 + 2 * half) * (2 * DK);
      bf.x = vr[dcol]; bf.y = vr[2 * DK + dcol];
    }
#pragma unroll 4
    for (int ks = 0; ks < 64; ++ks) {
      const int kk = khalf2 * 256 + ks * 4 + 2 * half;
      if (ks < 63) {
        const float* vr = Vb + (size_t)(c0 + kk + 4) * (2 * DK);
        bfn.x = vr[dcol]; bfn.y = vr[2 * DK + dcol];
      }
      v2f a; a.x = ch[row * CH_LD + kk]; a.y = ch[row * CH_LD + kk + 1];
      aacc = wmma4(a, bf, aacc);
      bf = bfn;
    }
    __syncthreads();
  }

  {
#pragma unroll
    for (int i = 0; i < 8; ++i)
      part[khalf2 * (16 * DK) + (i + 8 * half) * DK + dcol] = aacc[i];
  }
  __syncthreads();
  for (int e = tid; e < 16 * DK; e += 256) {
    const int r = e >> 6, d = e & 63;
    attn_out[(size_t)(b * S_LEN + q0 + r) * DMODEL + h * DK + d] =
        part[e] + part[16 * DK + e];
  }
}

// ---------------------------------------------------------------------------
extern "C" void kernel_launch(void* const* d_in, const int* in_sizes, int n_in,
                              void* d_out, int out_size, void* d_ws, size_t ws_size,
                              hipStream_t stream) {
  const float* Q    = (const float*)d_in[0];
  const float* K    = (const float*)d_in[1];
  // d_in[2] (V) unused by the reference (MQA derives Vs from K @ WKV^T)
  const float* WQw  = (const float*)d_in[3];
  const float* WQb  = (const float*)d_in[4];
  const float* WKVw = (const float*)d_in[5];
  const float* WKVb = (const float*)d_in[6];
  const float* Dw   = (const float*)d_in[7];
  const float* Db   = (const float*)d_in[8];

  float* out     = (float*)d_out;
  float* weights = out + (size_t)NB * S_LEN * DMODEL;

  float* Qp   = (float*)d_ws;
  float* KVp  = Qp  + (size_t)NROWS * DMODEL;
  float* attn = KVp + (size_t)NROWS * (2 * DK);

  const dim3 blk(256);

  gemm_bias_kernel<<<dim3(NROWS / 128, DMODEL / 64), blk, 0, stream>>>(
      Q, WQw, WQb, Qp, NROWS, DMODEL, DMODEL);

  gemm_bias_kernel<<<dim3(NROWS / 128, (2 * DK) / 64), blk, 0, stream>>>(
      K, WKVw, WKVb, KVp, NROWS, 2 * DK, DMODEL);

  mqa_attn_kernel<<<dim3(NB * NHEAD * (S_LEN / 16)), blk, 0, stream>>>(
      Qp, KVp, weights, attn);

  gemm_bias_kernel<<<dim3(NROWS / 128, DMODEL / 64), blk, 0, stream>>>(
      attn, Dw, Db, out, NROWS, DMODEL, DMODEL);
}